// GCNJaccard_Aug_75076028334520
// MI455X (gfx1250) — compile-verified
//
#include <hip/hip_runtime.h>
#include <hip/hip_bf16.h>
#include <stdint.h>

// Problem constants (match reference).
#define NN 4096      // nodes
#define DD 4096      // input features
#define HH 256       // hidden
#define CC 16        // classes
#define WORDS 128    // NN / 32 bitmask words per row

typedef float v2f __attribute__((ext_vector_type(2)));
typedef float v8f __attribute__((ext_vector_type(8)));

// ---------------------------------------------------------------- zeroing
__global__ void k_zero(unsigned int* p, int n) {
    int i = blockIdx.x * blockDim.x + threadIdx.x;
    if (i < n) p[i] = 0u;
}

// ------------------------------------------------- scatter bits (transposed)
// Set bit [col][row] so that row i of the mask == column i of the matrix,
// which is exactly the gather pattern of An's row i (An[i,j] ~ A[j,i]).
__global__ void k_scatter(const int* __restrict__ rows,
                          const int* __restrict__ cols,
                          int nnz, unsigned int* __restrict__ bits) {
    int k = blockIdx.x * blockDim.x + threadIdx.x;
    if (k >= nnz) return;
    int r = rows[k];
    int c = cols[k];
    atomicOr(&bits[(size_t)c * WORDS + (r >> 5)], 1u << (r & 31));
}

// ------------------------------- combine (a&jm)|ja|diag, popcount -> dinv
__global__ void k_combine(unsigned int* __restrict__ a,
                          const unsigned int* __restrict__ jm,
                          const unsigned int* __restrict__ ja,
                          float* __restrict__ dinv) {
    __shared__ unsigned int cnt[WORDS];
    int i = blockIdx.x;       // row of A^T == column of A
    int t = threadIdx.x;      // 128 threads, one word each
    size_t idx = (size_t)i * WORDS + t;
    unsigned int w = (a[idx] & jm[idx]) | ja[idx];
    if (t == (i >> 5)) w |= 1u << (i & 31);   // self loop: A[i,i] = 1
    a[idx] = w;                                // in-place: a becomes A_aug^T
    cnt[t] = __popc(w);
    __syncthreads();
    for (int s = 64; s > 0; s >>= 1) {
        if (t < s) cnt[t] += cnt[t + s];
        __syncthreads();
    }
    if (t == 0) {
        float deg = (float)cnt[0];             // deg = col-sum of A = row popcount of A^T
        dinv[i] = deg > 0.f ? rsqrtf(deg) : 0.f;
    }
}

// ----------------------------------------------- transpose W1 [H,D]->[D,H]
__global__ void k_transpose_w1(const float* __restrict__ w1,
                               float* __restrict__ w1t) {
    int c = blockIdx.x;       // 0..DD-1
    int t = threadIdx.x;      // 0..HH-1
    w1t[(size_t)c * HH + t] = w1[(size_t)t * DD + c];
}

// ------------------------------ h[r,:] += W1^T[c,:] for each attr nonzero
__global__ void k_spmm_x(const int* __restrict__ rows,
                         const int* __restrict__ cols,
                         const float* __restrict__ w1t,
                         float* __restrict__ h) {
    int k = blockIdx.x;       // one block per nnz
    int t = threadIdx.x;      // 256 threads = hidden dim
    int r = rows[k];
    int c = cols[k];
#if defined(__HIP_DEVICE_COMPILE__)
    unsafeAtomicAdd(&h[(size_t)r * HH + t], w1t[(size_t)c * HH + t]);
#else
    (void)r; (void)c;
#endif
}

// ---------------------- layer-1 propagate: h1 = relu(Dinv A^T Dinv h)
__global__ void k_prop1(const unsigned int* __restrict__ adjT,
                        const float* __restrict__ dinv,
                        const float* __restrict__ h,
                        float* __restrict__ h1) {
    __shared__ unsigned int row[WORDS];
    int i = blockIdx.x;       // output row
    int t = threadIdx.x;      // 256 threads = hidden column
    if (t < WORDS) row[t] = adjT[(size_t)i * WORDS + t];
    __syncthreads();
    float acc = 0.f;
    for (int w = 0; w < WORDS; ++w) {
        unsigned int bits = row[w];
        while (bits) {
            int b = __ffs(bits) - 1;
            bits &= bits - 1;
            int j = (w << 5) + b;
            acc += dinv[j] * h[(size_t)j * HH + t];
        }
    }
    float v = dinv[i] * acc;
    h1[(size_t)i * HH + t] = v > 0.f ? v : 0.f;
}

// ----------------------- layer-2 GEMM via WMMA f32: h2 = h1 @ W2^T
// One wave per 16-row tile; K=256 in steps of 4 with V_WMMA_F32_16X16X4_F32.
// A (16x4 f32): lanes 0-15 hold K=0,1; lanes 16-31 hold K=2,3 (VGPR0=K even).
// C/D (16x16 f32): VGPR v -> M = v + 8*(lane>=16), N = lane&15.
__global__ void k_gemm2(const float* __restrict__ h1,
                        const float* __restrict__ w2,
                        float* __restrict__ h2) {
    int tile = blockIdx.x;                 // 256 tiles of 16 rows
    int lane = threadIdx.x;                // 32 (wave32)
    int half = lane >> 4;
    int l    = lane & 15;
    const float* arow = h1 + ((size_t)(tile * 16 + l)) * HH + 2 * half;
    const float* brow = w2 + (size_t)l * HH + 2 * half;   // W2^T[k,n] = w2[n*HH+k]
    v8f acc = {};
    for (int kk = 0; kk < HH; kk += 4) {
        __builtin_prefetch(arow + kk + 64, 0, 0);
        v2f a; a.x = arow[kk]; a.y = arow[kk + 1];
        v2f b; b.x = brow[kk]; b.y = brow[kk + 1];
#if __has_builtin(__builtin_amdgcn_wmma_f32_16x16x4_f32)
        acc = __builtin_amdgcn_wmma_f32_16x16x4_f32(
            /*neg_a=*/false, a, /*neg_b=*/false, b,
            /*c_mod=*/(short)0, acc, /*reuse_a=*/false, /*reuse_b=*/false);
#else
        // Exact-f32 fallback (should not trigger on gfx1250 toolchains).
        for (int kq = 0; kq < 2; ++kq) {
            acc[0] += a[kq] * b[kq];
        }
#endif
    }
    int mbase = half * 8;
    for (int v = 0; v < 8; ++v) {
        h2[(size_t)(tile * 16 + mbase + v) * CC + l] = acc[v];
    }
}

// ---------------------- layer-2 propagate: out = Dinv A^T Dinv h2 + bias
__global__ void k_prop2(const unsigned int* __restrict__ adjT,
                        const float* __restrict__ dinv,
                        const float* __restrict__ h2,
                        const float* __restrict__ bias2,
                        float* __restrict__ out) {
    int t   = threadIdx.x;                 // 64 threads: 4 rows x 16 cols
    int sub = t >> 4;
    int c   = t & 15;
    int i   = blockIdx.x * 4 + sub;
    const unsigned int* row = adjT + (size_t)i * WORDS;
    float acc = 0.f;
    for (int w = 0; w < WORDS; ++w) {
        unsigned int bits = row[w];
        while (bits) {
            int b = __ffs(bits) - 1;
            bits &= bits - 1;
            int j = (w << 5) + b;
            acc += dinv[j] * h2[(size_t)j * CC + c];
        }
    }
    out[(size_t)i * CC + c] = dinv[i] * acc + bias2[c];
}

// ---------------------------------------------------------------- launch
extern "C" void kernel_launch(void* const* d_in, const int* in_sizes, int n_in,
                              void* d_out, int out_size, void* d_ws, size_t ws_size,
                              hipStream_t stream) {
    const int*   attr_row = (const int*)d_in[0];
    const int*   attr_col = (const int*)d_in[1];
    const int*   edge_idx = (const int*)d_in[2];   // [2, E] row-major
    const int*   jm_idx   = (const int*)d_in[3];   // [2, NNZ_JM]
    const int*   ja_idx   = (const int*)d_in[4];   // [2, NNZ_JA]
    const float* lin1_w   = (const float*)d_in[5]; // [H, D]
    const float* lin2_w   = (const float*)d_in[6]; // [C, H]
    const float* bias2    = (const float*)d_in[7]; // [C]
    float*       out      = (float*)d_out;         // [N, C]

    const int nnz_attr = in_sizes[0];
    const int E        = in_sizes[2] / 2;
    const int njm      = in_sizes[3] / 2;
    const int nja      = in_sizes[4] / 2;

    // Workspace layout (u32 units for offsets; everything 4B typed).
    const size_t BITS = (size_t)NN * WORDS;        // 524288 u32 per mask
    unsigned int* abitT  = (unsigned int*)d_ws;           // becomes A_aug^T
    unsigned int* jmbitT = abitT + BITS;
    unsigned int* jabitT = jmbitT + BITS;
    float* h    = (float*)(jabitT + BITS);                // [N, H] accum (needs zero)
    float* dinv = h + (size_t)NN * HH;                    // [N]
    float* w1t  = dinv + NN;                              // [D, H]
    float* h1   = w1t + (size_t)DD * HH;                  // [N, H]
    float* h2   = h1 + (size_t)NN * HH;                   // [N, C]

    // 1. zero bitmasks + h (contiguous region).
    {
        int nz = (int)(3 * BITS + (size_t)NN * HH);
        k_zero<<<(nz + 255) / 256, 256, 0, stream>>>((unsigned int*)d_ws, nz);
    }
    // 2. scatter transposed bitmasks.
    k_scatter<<<(E + 255) / 256, 256, 0, stream>>>(edge_idx, edge_idx + E, E, abitT);
    k_scatter<<<(njm + 255) / 256, 256, 0, stream>>>(jm_idx, jm_idx + njm, njm, jmbitT);
    k_scatter<<<(nja + 255) / 256, 256, 0, stream>>>(ja_idx, ja_idx + nja, nja, jabitT);
    // 3. combine + degree -> dinv.
    k_combine<<<NN, WORDS, 0, stream>>>(abitT, jmbitT, jabitT, dinv);
    // 4. W1^T, then sparse x @ W1^T with f32 atomics.
    k_transpose_w1<<<DD, HH, 0, stream>>>(lin1_w, w1t);
    k_spmm_x<<<nnz_attr, HH, 0, stream>>>(attr_row, attr_col, w1t, h);
    // 5. layer-1 propagate + relu.
    k_prop1<<<NN, HH, 0, stream>>>(abitT, dinv, h, h1);
    // 6. layer-2 GEMM on WMMA (one wave per 16-row tile).
    k_gemm2<<<NN / 16, 32, 0, stream>>>(h1, lin2_w, h2);
    // 7. layer-2 propagate + bias.
    k_prop2<<<NN / 4, 64, 0, stream>>>(abitT, dinv, h2, bias2, out);
}